// HeteroGraphCLF_15925738733689
// MI455X (gfx1250) — compile-verified
//
#include <hip/hip_runtime.h>

#define NN 62976        // nodes per type
#define EE 1007616      // edges per edge type
#define IND 246
#define HF 64           // hidden features
#define OC 192          // packed GEMM output cols: [W_lr | W_ll | sW]

typedef __attribute__((ext_vector_type(2))) float v2f;
typedef __attribute__((ext_vector_type(8))) float v8f;

// ---------------------------------------------------------------------------
// Pack [W_lr | W_ll | sW] (each [K,64]) into Wc [Kp,192], zero-padded rows.
// ---------------------------------------------------------------------------
__global__ void pack_w_kernel(const float* __restrict__ Wlr,
                              const float* __restrict__ Wll,
                              const float* __restrict__ sW,
                              float* __restrict__ Wc, int K, int Kp) {
    int idx = blockIdx.x * blockDim.x + threadIdx.x;
    if (idx >= Kp * OC) return;
    int k = idx / OC;
    int c = idx - k * OC;
    float v = 0.0f;
    if (k < K) {
        if (c < 64)       v = Wlr[k * 64 + c];
        else if (c < 128) v = Wll[k * 64 + (c - 64)];
        else              v = sW [k * 64 + (c - 128)];
    }
    Wc[idx] = v;
}

// ---------------------------------------------------------------------------
// FP32 WMMA GEMM:  out[N,192] = X[N,K] @ Wc[Kp,192]   (Kp multiple of 4, rows
// K..Kp-1 of Wc are zero).  Block = 256 threads = 8 waves, computes 32 rows.
// Wave (r,c): rows r*16..+15, col tiles {c, c+4, c+8} -> 3 v8f accumulators.
// Uses V_WMMA_F32_16X16X4_F32 (exact fp32 math, matches reference).
// A 16x4 f32 layout: lanes 0-15 hold K=(0,1) in (a.x,a.y), lanes 16-31 K=(2,3).
// C/D 16x16 f32: VGPR v, lane l -> M = v + 8*(l/16), N = l%16.
// ---------------------------------------------------------------------------
__global__ __launch_bounds__(256)
void gemm_wmma_kernel(const float* __restrict__ X, const float* __restrict__ Wc,
                      float* __restrict__ out, int K, int Kp) {
    __shared__ float As[32 * 252];            // max Kp=248, stride Kp+4
    const int tid  = threadIdx.x;
    const int row0 = blockIdx.x * 32;
    const int ls   = Kp + 4;                  // conflict-free LDS stride

    for (int idx = tid; idx < 32 * Kp; idx += 256) {
        int r = idx / Kp;
        int c = idx - r * Kp;
        As[r * ls + c] = (c < K) ? X[(size_t)(row0 + r) * K + c] : 0.0f;
    }
    __syncthreads();

    const int wave   = tid >> 5;
    const int lane   = tid & 31;
    const int lane16 = lane & 15;
    const int hi     = lane >> 4;             // half-wave id
    const int rhalf  = wave >> 2;             // 0/1: which 16-row half
    const int cgrp   = wave & 3;              // base col tile
    const int kHalf  = hi * 2;                // K sub-offset for this half-wave

    v8f c0 = {}; v8f c1 = {}; v8f c2 = {};
    const float* arow = &As[(rhalf * 16 + lane16) * ls];

    for (int k0 = 0; k0 < Kp; k0 += 4) {
        v2f a;
        a.x = arow[k0 + kHalf];
        a.y = arow[k0 + kHalf + 1];
        const float* b0 = &Wc[(size_t)(k0 + kHalf) * OC + lane16];
        v2f b;
        b.x = b0[cgrp * 16];
        b.y = b0[OC + cgrp * 16];
        c0 = __builtin_amdgcn_wmma_f32_16x16x4_f32(false, a, false, b, (short)0, c0, false, false);
        b.x = b0[(cgrp + 4) * 16];
        b.y = b0[OC + (cgrp + 4) * 16];
        c1 = __builtin_amdgcn_wmma_f32_16x16x4_f32(false, a, false, b, (short)0, c1, false, false);
        b.x = b0[(cgrp + 8) * 16];
        b.y = b0[OC + (cgrp + 8) * 16];
        c2 = __builtin_amdgcn_wmma_f32_16x16x4_f32(false, a, false, b, (short)0, c2, false, false);
    }

    const int mbase = row0 + rhalf * 16 + hi * 8;
#pragma unroll
    for (int v = 0; v < 8; ++v) {
        size_t ro = (size_t)(mbase + v) * OC + lane16;
        out[ro + cgrp * 16]       = c0[v];
        out[ro + (cgrp + 4) * 16] = c1[v];
        out[ro + (cgrp + 8) * 16] = c2[v];
    }
}

// ---------------------------------------------------------------------------
// In-degree per destination (input-invariant; computed once per call).
// ---------------------------------------------------------------------------
__global__ void deg_kernel(const int* __restrict__ dst, float* __restrict__ deg, int E) {
    int e = blockIdx.x * blockDim.x + threadIdx.x;
    if (e < E) atomicAdd(&deg[dst[e]], 1.0f);
}

// ---------------------------------------------------------------------------
// Edge scatter: acc[dst[e], h] += feat[src[e], featOff+h] * (w[e]*We[h]+be[h])
// One thread per (edge, feature); 64 consecutive threads cover one edge row.
// ---------------------------------------------------------------------------
__global__ __launch_bounds__(256)
void scatter_kernel(const float* __restrict__ feat, int featOff,
                    const int* __restrict__ src, const int* __restrict__ dst,
                    const float* __restrict__ w, const float* __restrict__ We,
                    const float* __restrict__ be, float* __restrict__ acc, int E) {
    int gid = blockIdx.x * 256 + threadIdx.x;
    int e = gid >> 6;
    if (e >= E) return;
    int h = gid & 63;
    float wf = w[e] * We[h] + be[h];
    float m  = feat[(size_t)src[e] * OC + featOff + h] * wf;
    atomicAdd(&acc[(size_t)dst[e] * HF + h], m);
}

// ---------------------------------------------------------------------------
// h = relu(0.5*(mean_inter + mean_intra)) + skip + sb
// mean_* = deg>0 ? acc/deg : 0 ;  skip lives in feat[:,128:192].
// ---------------------------------------------------------------------------
__global__ __launch_bounds__(256)
void combine_kernel(const float* __restrict__ acc_inter, const float* __restrict__ deg_inter,
                    const float* __restrict__ acc_intra, const float* __restrict__ deg_intra,
                    const float* __restrict__ feat, const float* __restrict__ sb,
                    float* __restrict__ hout, int N) {
    int gid = blockIdx.x * 256 + threadIdx.x;
    int i = gid >> 6;
    if (i >= N) return;
    int h = gid & 63;
    float di = deg_inter[i];
    float mi = (di > 0.0f) ? acc_inter[(size_t)i * HF + h] / di : 0.0f;
    float da = deg_intra[i];
    float ma = (da > 0.0f) ? acc_intra[(size_t)i * HF + h] / da : 0.0f;
    float r = 0.5f * (mi + ma);
    r = (r > 0.0f) ? r : 0.0f;
    hout[(size_t)i * HF + h] = r + feat[(size_t)i * OC + 128 + h] + sb[h];
}

// ---------------------------------------------------------------------------
extern "C" void kernel_launch(void* const* d_in, const int* in_sizes, int n_in,
                              void* d_out, int out_size, void* d_ws, size_t ws_size,
                              hipStream_t stream) {
    (void)in_sizes; (void)n_in; (void)out_size; (void)ws_size;

    const float* x_l = (const float*)d_in[0];
    const float* x_r = (const float*)d_in[1];
    const int*   src_ll = (const int*)d_in[2];
    const int*   dst_ll = (const int*)d_in[3];
    const float* w_ll   = (const float*)d_in[4];
    const int*   src_rr = (const int*)d_in[5];
    const int*   dst_rr = (const int*)d_in[6];
    const float* w_rr   = (const float*)d_in[7];
    const int*   src_lr = (const int*)d_in[8];
    const int*   dst_lr = (const int*)d_in[9];
    const float* w_lr   = (const float*)d_in[10];
    const int*   src_rl = (const int*)d_in[11];
    const int*   dst_rl = (const int*)d_in[12];
    const float* w_rl   = (const float*)d_in[13];
    // params, insertion order per layer: W_lr, We_lr, be_lr, W_ll, We_ll, be_ll, sW, sb
    const int PBASE = 14;

    // ---- workspace layout (floats) ----
    float* ws = (float*)d_ws;
    float* feat_l = ws;                  ws += (size_t)NN * OC;
    float* feat_r = ws;                  ws += (size_t)NN * OC;
    float* acc_base = ws;
    float* acc_inter_r = acc_base + 0 * (size_t)NN * HF;
    float* acc_inter_l = acc_base + 1 * (size_t)NN * HF;
    float* acc_intra_l = acc_base + 2 * (size_t)NN * HF;
    float* acc_intra_r = acc_base + 3 * (size_t)NN * HF;
    ws += 4 * (size_t)NN * HF;
    float* deg_base = ws;
    float* deg_ll = deg_base + 0 * (size_t)NN;
    float* deg_rr = deg_base + 1 * (size_t)NN;
    float* deg_lr = deg_base + 2 * (size_t)NN;
    float* deg_rl = deg_base + 3 * (size_t)NN;
    ws += 4 * (size_t)NN;
    float* Wc = ws;                      // up to 248*192

    float* hL = (float*)d_out;
    float* hR = hL + (size_t)NN * HF;

    // ---- degrees (shared by both layers) ----
    hipMemsetAsync(deg_base, 0, 4 * (size_t)NN * sizeof(float), stream);
    {
        dim3 g((EE + 255) / 256), b(256);
        deg_kernel<<<g, b, 0, stream>>>(dst_ll, deg_ll, EE);
        deg_kernel<<<g, b, 0, stream>>>(dst_rr, deg_rr, EE);
        deg_kernel<<<g, b, 0, stream>>>(dst_lr, deg_lr, EE);
        deg_kernel<<<g, b, 0, stream>>>(dst_rl, deg_rl, EE);
    }

    for (int layer = 0; layer < 2; ++layer) {
        const float* W_lr = (const float*)d_in[PBASE + layer * 8 + 0];
        const float* We_l = (const float*)d_in[PBASE + layer * 8 + 1];
        const float* be_l = (const float*)d_in[PBASE + layer * 8 + 2];
        const float* W_ll = (const float*)d_in[PBASE + layer * 8 + 3];
        const float* We_i = (const float*)d_in[PBASE + layer * 8 + 4];
        const float* be_i = (const float*)d_in[PBASE + layer * 8 + 5];
        const float* sW   = (const float*)d_in[PBASE + layer * 8 + 6];
        const float* sb   = (const float*)d_in[PBASE + layer * 8 + 7];

        const int K  = (layer == 0) ? IND : HF;
        const int Kp = (layer == 0) ? 248 : 64;
        const float* Xl = (layer == 0) ? x_l : hL;
        const float* Xr = (layer == 0) ? x_r : hR;

        pack_w_kernel<<<(Kp * OC + 255) / 256, 256, 0, stream>>>(W_lr, W_ll, sW, Wc, K, Kp);
        gemm_wmma_kernel<<<NN / 32, 256, 0, stream>>>(Xl, Wc, feat_l, K, Kp);
        gemm_wmma_kernel<<<NN / 32, 256, 0, stream>>>(Xr, Wc, feat_r, K, Kp);

        hipMemsetAsync(acc_base, 0, 4 * (size_t)NN * HF * sizeof(float), stream);

        dim3 sg(((size_t)EE * 64 + 255) / 256), sb256(256);
        // inter_r: conv(x_l, W_lr, ...) over lr edges
        scatter_kernel<<<sg, sb256, 0, stream>>>(feat_l, 0,  src_lr, dst_lr, w_lr, We_l, be_l, acc_inter_r, EE);
        // inter_l: conv(x_r, W_lr, ...) over rl edges
        scatter_kernel<<<sg, sb256, 0, stream>>>(feat_r, 0,  src_rl, dst_rl, w_rl, We_l, be_l, acc_inter_l, EE);
        // intra_l: conv(x_l, W_ll, ...) over ll edges
        scatter_kernel<<<sg, sb256, 0, stream>>>(feat_l, 64, src_ll, dst_ll, w_ll, We_i, be_i, acc_intra_l, EE);
        // intra_r: conv(x_r, W_ll, ...) over rr edges
        scatter_kernel<<<sg, sb256, 0, stream>>>(feat_r, 64, src_rr, dst_rr, w_rr, We_i, be_i, acc_intra_r, EE);

        dim3 cg(((size_t)NN * HF + 255) / 256);
        combine_kernel<<<cg, sb256, 0, stream>>>(acc_inter_l, deg_rl, acc_intra_l, deg_ll, feat_l, sb, hL, NN);
        combine_kernel<<<cg, sb256, 0, stream>>>(acc_inter_r, deg_lr, acc_intra_r, deg_rr, feat_r, sb, hR, NN);
    }
}